// PulseTrainSynth_71201967833414
// MI455X (gfx1250) — compile-verified
//
#include <hip/hip_runtime.h>
#include <stdint.h>

#define NC 8
#define NH 64
#define NBANDS 32
#define BATCH 2
#define TLEN 65536
#define TWO_PI_F 6.283185307179586f
#define PI_F 3.141592653589793f
#define SRATE 48000.0f

// ---------------- workspace layout (float element offsets) ----------------
#define WS_BANKA   0
#define WS_BANKB   (WS_BANKA + BATCH*TLEN)
#define WS_RESA    (WS_BANKB + BATCH*TLEN)
#define WS_RESB    (WS_RESA  + BATCH*TLEN)
#define WS_PHI     (WS_RESB  + BATCH*TLEN)
#define WS_CSUM    (WS_PHI   + BATCH*TLEN)
#define WS_COFF    (WS_CSUM  + 64)
#define WS_COEF    (WS_COFF  + 64)

#define CHUNK 2048
#define NCHUNK (TLEN/CHUNK)   // 32 chunks per batch

typedef __attribute__((ext_vector_type(2))) float v2f;
typedef __attribute__((ext_vector_type(8))) float v8f;

#if defined(__gfx1250__) && __has_builtin(__builtin_amdgcn_wmma_f32_16x16x4_f32)
#define HAVE_WMMA_F32X4 1
#endif

// ---------------- K1: per-chunk partial sums of pi*f0/SR ----------------
__global__ void k_chunksum(const float* __restrict__ f0, float* ws) {
    __shared__ float red[256];
    int gid = blockIdx.x;                    // 0..63
    int b = gid / NCHUNK, ch = gid % NCHUNK;
    const float* x = f0 + (size_t)b*TLEN + (size_t)ch*CHUNK;
    float s = 0.f;
    for (int i = threadIdx.x; i < CHUNK; i += 256) s += x[i] * (PI_F / SRATE);
    red[threadIdx.x] = s; __syncthreads();
    for (int st = 128; st > 0; st >>= 1) {
        if ((int)threadIdx.x < st) red[threadIdx.x] += red[threadIdx.x + st];
        __syncthreads();
    }
    if (threadIdx.x == 0) ws[WS_CSUM + gid] = red[0];
}

// ---------------- K2: exclusive scan of chunk sums (per batch) ----------------
__global__ void k_chunkoff(float* ws) {
    int b = threadIdx.x;
    if (b >= BATCH) return;
    float run = 0.f;
    for (int i = 0; i < NCHUNK; ++i) {
        float v = ws[WS_CSUM + b*NCHUNK + i];
        ws[WS_COFF + b*NCHUNK + i] = run;
        run += v;
    }
}

// ---------------- K3: phase prefix-scan; 16-wide segments via WMMA ----------------
// Each wave scans 256 contiguous samples as a 16(seg) x 16(pos) tile:
// Y = L16 * X done as 4 chained V_WMMA_F32_16X16X4_F32 (K blocks of 4).
__global__ void k_scan(const float* __restrict__ f0, float* ws) {
    __shared__ float sx[CHUNK];
    __shared__ float sy[CHUNK];
    __shared__ float ssum[128];
    int gid = blockIdx.x;
    int b = gid / NCHUNK, ch = gid % NCHUNK;
    size_t base = (size_t)b*TLEN + (size_t)ch*CHUNK;
    for (int i = threadIdx.x; i < CHUNK; i += 256) sx[i] = f0[base + i] * (PI_F / SRATE);
    __syncthreads();

    int w = threadIdx.x >> 5;        // wave id (0..7), wave32
    int lane = threadIdx.x & 31;
    int m = lane & 15;               // A row / B,D column
    int h = lane >> 4;               // lane half
    const float* wx = sx + w * 256;

#ifdef HAVE_WMMA_F32X4
    v8f acc = {0.f,0.f,0.f,0.f,0.f,0.f,0.f,0.f};
#pragma unroll
    for (int q = 0; q < 4; ++q) {
        int j0 = 4*q + 2*h, j1 = j0 + 1;
        v2f a, bb;
        a[0] = (j0 <= m) ? 1.f : 0.f;          // A = lower-triangular ones (16x4 slice)
        a[1] = (j1 <= m) ? 1.f : 0.f;
        bb[0] = wx[m*16 + j0];                 // B = X[4q:4q+4, :], col = m
        bb[1] = wx[m*16 + j1];
        acc = __builtin_amdgcn_wmma_f32_16x16x4_f32(
                  false, a, false, bb, (short)0, acc, false, false);
    }
    if (h == 1) ssum[w*16 + m] = acc[7];       // row 15 = segment inclusive sum
#else
    float accA[8];
    {
        float run = 0.f;
        for (int j = 0; j < 8*h; ++j) run += wx[m*16 + j];
        for (int r = 0; r < 8; ++r) { run += wx[m*16 + 8*h + r]; accA[r] = run; }
    }
    if (h == 1) ssum[w*16 + m] = accA[7];
#endif
    __syncthreads();
    if (threadIdx.x == 0) {                    // exclusive scan of 128 segment sums
        float run = ws[WS_COFF + gid];         // seed with cross-chunk offset
        for (int i = 0; i < 128; ++i) { float v = ssum[i]; ssum[i] = run; run += v; }
    }
    __syncthreads();
    float off = ssum[w*16 + m];
#pragma unroll
    for (int r = 0; r < 8; ++r) {
#ifdef HAVE_WMMA_F32X4
        sy[w*256 + m*16 + 8*h + r] = acc[r] + off;
#else
        sy[w*256 + m*16 + 8*h + r] = accA[r] + off;
#endif
    }
    __syncthreads();
    for (int i = threadIdx.x; i < CHUNK; i += 256) ws[WS_PHI + base + i] = sy[i];
}

// ---------------- K4: main synth (bandwidth-bound stream over amps) ----------------
__global__ void k_main(const float* __restrict__ hamps,
                       const float* __restrict__ hfm,
                       const float* __restrict__ cps,
                       const float* __restrict__ nbamp,
                       const float* __restrict__ nps,
                       const float* __restrict__ npg_,
                       const float* __restrict__ fng_,
                       const float* __restrict__ bands,
                       float* ws) {
    int g = blockIdx.x * 256 + threadIdx.x;       // 0 .. B*T-1
    int b = g / TLEN, t = g % TLEN;
    float phi = ws[WS_PHI + g];
    float fm  = hfm[g];

    // noise = sum_n nbamp[b,n,t] * bands[n,t]
    float noise = 0.f;
#pragma unroll 4
    for (int n = 0; n < NBANDS; ++n)
        noise = fmaf(nbamp[(size_t)(b*NBANDS + n)*TLEN + t],
                     bands[(size_t)n*TLEN + t], noise);

    float nphase = fmodf(phi, TWO_PI_F);
    float na  = nps[(size_t)(b*2 + 0)*TLEN + t];
    float nbe = nps[(size_t)(b*2 + 1)*TLEN + t];
    float nenv = (1.f - __expf(-na*nphase)) * __expf(-nbe*nphase);
    float npg = npg_[g], fng = fng_[g];
    float tf   = noise * (npg + fng) * 0.7f;               // turbulence factor
    float addc = noise*nenv*npg + noise*fng*0.3f;          // per-cylinder additive

    const int forder[NC] = {0,4,3,7,5,2,6,1};
    float sumA = 0.f, sumB = 0.f;
#pragma unroll 1
    for (int c = 0; c < NC; ++c) {
        float ang = (float)forder[c] * (TWO_PI_F / (float)NC);
        float cph = fmodf(phi + ang, TWO_PI_F);
        float hp  = powf(cph * (1.f/TWO_PI_F), fm) * TWO_PI_F;
        float s, cs; sincosf(hp, &s, &cs);
        float twoc = 2.f * cs;
        float sp = 0.f, scur = s, accH = 0.f;
        const float* ap = hamps + (size_t)(b*NC*NH + c*NH)*TLEN + t;
        if (c + 1 < NC) __builtin_prefetch(ap + (size_t)NH*TLEN, 0, 0);
#pragma unroll 8
        for (int hh = 0; hh < NH; ++hh) {
            float amp = ap[(size_t)hh*TLEN];
            accH = fmaf(amp, -scur, accH);                 // osc = -sin((hh+1)*hp)
            float sn = fmaf(twoc, scur, -sp);              // Chebyshev recurrence
            sp = scur; scur = sn;
        }
        float alpha = cps[(size_t)(b*2*NC + c)*TLEN + t];
        float beta  = cps[(size_t)(b*2*NC + NC + c)*TLEN + t];
        float env = (1.f - __expf(-alpha*cph)) * __expf(-beta*cph);
        float harm = accH * env * 10.f;
        float cyl = harm * (1.f + tf) + addc;
        if (c < 4) sumA += cyl; else sumB += cyl;
    }
    ws[WS_BANKA + g] = sumA;
    ws[WS_BANKB + g] = sumB;
}

// ---------------- K5: resonator coefficients (Threefry-2x32-20 gumbel argmax) ----------------
__device__ inline uint32_t rotl32(uint32_t x, int d){ return (x<<d)|(x>>(32-d)); }
__device__ void tf2x32(uint32_t k0, uint32_t k1, uint32_t x0, uint32_t x1,
                       uint32_t& o0, uint32_t& o1) {
    const int R0[4] = {13,15,26,6};
    const int R1[4] = {17,29,16,24};
    uint32_t ks[3] = {k0, k1, 0x1BD11BDAu ^ k0 ^ k1};
    x0 += ks[0]; x1 += ks[1];
    for (int gr = 0; gr < 5; ++gr) {
        const int* r = (gr & 1) ? R1 : R0;
        for (int i = 0; i < 4; ++i) { x0 += x1; x1 = rotl32(x1, r[i]); x1 ^= x0; }
        x0 += ks[(gr+1)%3];
        x1 += ks[(gr+2)%3] + (uint32_t)(gr+1);
    }
    o0 = x0; o1 = x1;
}
__device__ inline float gumbel_from_bits(uint32_t bits) {
    float u = __uint_as_float((bits >> 9) | 0x3F800000u) - 1.0f;   // [0,1)
    u = fmaxf(u, 1.1754944e-38f);
    return -logf(-logf(u));
}
__global__ void k_coefs(const float* da, const float* fa, const float* ra,
                        const float* db, const float* fb, const float* rb,
                        const float* dc, const float* fc, const float* rc,
                        float* ws) {
    int r = threadIdx.x;
    if (r >= 3) return;
    // jax.random.key(42) -> split(3): keys from threefry(key, iota(6))
    uint32_t p00,p01,p10,p11,p20,p21;
    tf2x32(0u,42u, 0u,3u, p00,p01);
    tf2x32(0u,42u, 1u,4u, p10,p11);
    tf2x32(0u,42u, 2u,5u, p20,p21);
    uint32_t K0, K1;
    if (r == 0)      { K0 = p00; K1 = p10; }   // ka
    else if (r == 1) { K0 = p20; K1 = p01; }   // kb
    else             { K0 = p11; K1 = p21; }   // kc
    const float* logits = (r==0)?da:(r==1)?db:dc;
    const float* fbg    = (r==0)?fa:(r==1)?fb:fc;
    const float* refl   = (r==0)?ra:(r==1)?rb:rc;
    int L    = (r==2)? 8 : 85;
    int minD = (r==2)? 32 : 40;
    bool lowpass = (r==2);

    int half = (L + 1) / 2;
    float best = -3.4e38f; int kbest = 0;
    for (int j = 0; j < L; ++j) {
        uint32_t x0, x1; bool second;
        if (j < half) { x0 = (uint32_t)j; x1 = (half+j < L) ? (uint32_t)(half+j) : 0u; second = false; }
        else          { x0 = (uint32_t)(j - half); x1 = (uint32_t)j; second = true; }
        uint32_t y0, y1; tf2x32(K0, K1, x0, x1, y0, y1);
        float v = logits[j] + gumbel_from_bits(second ? y1 : y0);
        if (v > best) { best = v; kbest = j; }
    }
    float r0 = refl[0], r1 = refl[1];
    if (lowpass) { r0 = 1.f/(1.f+expf(-r0)); r1 = 1.f/(1.f+expf(-r1)); }
    else         { r0 = tanhf(r0);           r1 = tanhf(r1); }
    float k1v = tanhf(r0);                       // resonant_activation, tau = 0
    float k2v = tanhf(r1);
    float a1 = k1v * (1.f - k2v);
    float a2 = fminf(fmaxf(k2v, -0.999f), 0.999f);
    float a1b = 0.999f - fabsf(a2);
    a1 = fminf(fmaxf(a1, -a1b), a1b);
    float gg = powf(1.f/(1.f+expf(-fbg[0])), 0.45f);
    int D1 = minD + kbest + 1;
    int D2 = minD + ((kbest + 1) % L) + 1;       // jnp.roll wraparound
    ws[WS_COEF + r*4 + 0] = a1 * gg;
    ws[WS_COEF + r*4 + 1] = a2 * gg;
    ws[WS_COEF + r*4 + 2] = (float)D1;
    ws[WS_COEF + r*4 + 3] = (float)D2;
}

// ---------------- K6: resonators a & b (2-tap comb IIR, history in LDS) ----------------
__global__ void k_reso_ab(float* ws) {
    extern __shared__ float yl[];                // TLEN floats = 256 KB (<= 320 KB/WGP)
    int which = blockIdx.x >> 1;                 // 0 = a, 1 = b
    int b = blockIdx.x & 1;
    const float* x = ws + (which == 0 ? WS_BANKA : WS_BANKB) + (size_t)b*TLEN;
    float* yo      = ws + (which == 0 ? WS_RESA  : WS_RESB ) + (size_t)b*TLEN;
    float A1 = ws[WS_COEF + which*4 + 0];
    float A2 = ws[WS_COEF + which*4 + 1];
    int D1 = (int)ws[WS_COEF + which*4 + 2];
    int D2 = (int)ws[WS_COEF + which*4 + 3];
    int W = min(D1, D2); if (W > (int)blockDim.x) W = blockDim.x;
    for (int t0 = 0; t0 < TLEN; t0 += W) {
        int t = t0 + (int)threadIdx.x;
        if ((int)threadIdx.x < W && t < TLEN) {
            float y = x[t];
            if (t >= D1) y = fmaf(-A1, yl[t - D1], y);
            if (t >= D2) y = fmaf(-A2, yl[t - D2], y);
            yl[t] = y;
        }
        __syncthreads();
    }
    for (int i = threadIdx.x; i < TLEN; i += blockDim.x) yo[i] = yl[i];
}

// ---------------- K7: resonator c on (resA + resB), write output ----------------
__global__ void k_reso_c(float* ws, float* __restrict__ out) {
    extern __shared__ float yl[];
    int b = blockIdx.x;
    const float* xa = ws + WS_RESA + (size_t)b*TLEN;
    const float* xb = ws + WS_RESB + (size_t)b*TLEN;
    float A1 = ws[WS_COEF + 8 + 0];
    float A2 = ws[WS_COEF + 8 + 1];
    int D1 = (int)ws[WS_COEF + 8 + 2];
    int D2 = (int)ws[WS_COEF + 8 + 3];
    int W = min(D1, D2); if (W > (int)blockDim.x) W = blockDim.x;
    for (int t0 = 0; t0 < TLEN; t0 += W) {
        int t = t0 + (int)threadIdx.x;
        if ((int)threadIdx.x < W && t < TLEN) {
            float y = xa[t] + xb[t];
            if (t >= D1) y = fmaf(-A1, yl[t - D1], y);
            if (t >= D2) y = fmaf(-A2, yl[t - D2], y);
            yl[t] = y;
        }
        __syncthreads();
    }
    for (int i = threadIdx.x; i < TLEN; i += blockDim.x)
        out[(size_t)b*TLEN + i] = yl[i];
}

extern "C" void kernel_launch(void* const* d_in, const int* in_sizes, int n_in,
                              void* d_out, int out_size, void* d_ws, size_t ws_size,
                              hipStream_t stream) {
    (void)in_sizes; (void)n_in; (void)out_size; (void)ws_size;
    const float* f0    = (const float*)d_in[0];
    const float* hamps = (const float*)d_in[1];
    const float* hfm   = (const float*)d_in[2];
    /* d_in[3] cylinder_phase_offsets: only its shape is used by the reference */
    const float* cps   = (const float*)d_in[4];
    const float* nbamp = (const float*)d_in[5];
    const float* nps   = (const float*)d_in[6];
    const float* npg   = (const float*)d_in[7];
    const float* fng   = (const float*)d_in[8];
    const float* bands = (const float*)d_in[9];
    const float* da = (const float*)d_in[10];
    const float* fa = (const float*)d_in[11];
    const float* ra = (const float*)d_in[12];
    const float* db = (const float*)d_in[13];
    const float* fb = (const float*)d_in[14];
    const float* rb = (const float*)d_in[15];
    const float* dc = (const float*)d_in[16];
    const float* fc = (const float*)d_in[17];
    const float* rc = (const float*)d_in[18];
    float* ws  = (float*)d_ws;
    float* out = (float*)d_out;

    k_chunksum<<<BATCH*NCHUNK, 256, 0, stream>>>(f0, ws);
    k_chunkoff<<<1, 32, 0, stream>>>(ws);
    k_scan    <<<BATCH*NCHUNK, 256, 0, stream>>>(f0, ws);
    k_main    <<<(BATCH*TLEN)/256, 256, 0, stream>>>(hamps, hfm, cps, nbamp, nps,
                                                     npg, fng, bands, ws);
    k_coefs   <<<1, 32, 0, stream>>>(da, fa, ra, db, fb, rb, dc, fc, rc, ws);
    k_reso_ab <<<4, 128, TLEN*sizeof(float), stream>>>(ws);
    k_reso_c  <<<BATCH, 128, TLEN*sizeof(float), stream>>>(ws, out);
}